// Router_37623913513629
// MI455X (gfx1250) — compile-verified
//
#include <hip/hip_runtime.h>
#include <hip/hip_bf16.h>
#include <math.h>

// ---------------------------------------------------------------------------
// Problem constants (from reference): N=8192 tokens, C=64 ch, H=W=8, E=16,
// Ccap = ceil(1.25*8192/16) = 640.
// Output layout (flat f32): dispatch[N*E*Ccap] | combine[N*E*Ccap] |
//                           z_loss | aux_loss | logits_std | logits[N*E]
// ---------------------------------------------------------------------------
#define NTOK   8192
#define NCH    64
#define NEXP   16
#define CCAP   640
#define HW     64           // H*W

static constexpr size_t COMB_OFF = (size_t)NTOK * NEXP * CCAP;   // 83,886,080
static constexpr size_t ZLOSS_OFF = 2 * COMB_OFF;                // 167,772,160
static constexpr size_t LOGIT_OFF = ZLOSS_OFF + 3;

// workspace layout (floats unless noted)
#define WS_Z     0          // sum of lse^2
#define WS_LSUM  1          // sum of logits
#define WS_LSQ   2          // sum of logits^2
#define WS_PSUM  3          // [16] sum of probs per expert
#define WS_MCNT  19         // [16] surviving-token count per expert
#define WS_PROB  64         // [8192] top-1 prob per token
#define WS_EXP   (64+NTOK)  // [8192] argmax expert per token (int)

typedef __attribute__((ext_vector_type(2))) float v2f;
typedef __attribute__((ext_vector_type(4))) float v4f;
typedef __attribute__((ext_vector_type(8))) float v8f;

// ---------------------------------------------------------------------------
// 0) zero the small accumulator region of the workspace
// ---------------------------------------------------------------------------
__global__ void init_ws(float* ws) {
    int t = threadIdx.x;
    if (t < 35) ws[t] = 0.0f;
}

// ---------------------------------------------------------------------------
// 1) streaming non-temporal zero of dispatch+combine (671 MB — the roofline)
// ---------------------------------------------------------------------------
__global__ void zero_big(v4f* __restrict__ p, long n4) {
    long i = (long)blockIdx.x * blockDim.x + threadIdx.x;
    long stride = (long)gridDim.x * blockDim.x;
    v4f z = {0.f, 0.f, 0.f, 0.f};
    for (; i < n4; i += stride) {
        __builtin_nontemporal_store(z, &p[i]);
    }
}

// ---------------------------------------------------------------------------
// 2) fused pool -> WMMA gating GEMM -> softmax/argmax/lse
//    block = 128 threads (4 waves), 64 tokens per block, grid = 128 blocks.
//    Each wave WMMAs a 16-token x 16-expert tile over K=64 with
//    V_WMMA_F32_16X16X4_F32 (16 chained MMAs).
// ---------------------------------------------------------------------------
#define LDSP 68   // padded pooled row stride: bank = 4*lane mod 64, conflict-free

__global__ void pool_gate(const float* __restrict__ X,
                          const float* __restrict__ Wg,
                          const float* __restrict__ bg,
                          float* __restrict__ out,
                          float* __restrict__ ws) {
    __shared__ float ldsPooled[64 * LDSP];
    __shared__ float ldsWg[NCH * NEXP];

    const int tid  = threadIdx.x;
    const int n0   = blockIdx.x * 64;            // first token of block

    // --- load gate weights to LDS (1024 f32) ---
    for (int i = tid; i < NCH * NEXP; i += 128) ldsWg[i] = Wg[i];

    // --- global average pool: 64 tokens x 64 channels, 64 f32 each ---
    const float* Xb = X + (size_t)n0 * (NCH * HW);
    for (int idx = tid; idx < 64 * NCH; idx += 128) {
        int t = idx >> 6, c = idx & 63;
        const v4f* p = (const v4f*)(Xb + (size_t)t * (NCH * HW) + c * HW);
        float s = 0.f;
#pragma unroll
        for (int q = 0; q < HW / 4; ++q) {
            v4f v = p[q];
            s += v.x + v.y + v.z + v.w;
        }
        ldsPooled[t * LDSP + c] = s * (1.0f / 64.0f);
    }
    __syncthreads();

    // --- WMMA gating GEMM: one wave per 16-token tile ---
    const int wave = tid >> 5;
    const int lane = tid & 31;
    const int half = lane >> 4;       // 0: lanes 0-15, 1: lanes 16-31
    const int lr   = lane & 15;       // expert index / row-in-half
    const int tb   = wave * 16;       // tile token base within block

    v8f acc = {};
#pragma unroll
    for (int kk = 0; kk < 16; ++kk) {
        const int k0 = kk * 4 + half * 2;   // K offsets {0,1} / {2,3}
        v2f a, b;
        a.x = ldsPooled[(tb + lr) * LDSP + k0];       // A[M=lr][K=k0]
        a.y = ldsPooled[(tb + lr) * LDSP + k0 + 1];   // A[M=lr][K=k0+1]
        b.x = ldsWg[(k0)     * NEXP + lr];            // B[K=k0][N=lr]
        b.y = ldsWg[(k0 + 1) * NEXP + lr];            // B[K=k0+1][N=lr]
        acc = __builtin_amdgcn_wmma_f32_16x16x4_f32(
                  false, a, false, b, (short)0, acc, false, false);
    }

    const float bias = bg[lr];

    float* wsProb = ws + WS_PROB;
    int*   wsExp  = (int*)ws + WS_EXP;

    float pacc = 0.f;   // per-expert prob sum (this lane's expert lr)
    float sacc = 0.f;   // logit sum
    float qacc = 0.f;   // logit^2 sum
    float zacc = 0.f;   // lse^2 sum (valid on lr==0 lanes)

#pragma unroll
    for (int r = 0; r < 8; ++r) {
        const int token = n0 + tb + r + half * 8;     // C/D layout: M = r + 8*half
        float logit = (acc[r] + bias) * (1.0f / 1.5f);
        logit = fminf(10.0f, fmaxf(-10.0f, logit));

        // row-wise (16-lane half) max + argmax, first-index tie-break
        float mval = logit; int me = lr;
#pragma unroll
        for (int m = 1; m <= 8; m <<= 1) {
            float ov = __shfl_xor(mval, m, 32);
            int   oe = __shfl_xor(me,   m, 32);
            if (ov > mval || (ov == mval && oe < me)) { mval = ov; me = oe; }
        }
        // sum of exp(logit - max)
        float ex = __expf(logit - mval);
        float s = ex;
#pragma unroll
        for (int m = 1; m <= 8; m <<= 1) s += __shfl_xor(s, m, 32);

        const float prob = ex / s;            // this lane's expert prob
        pacc += prob;
        sacc += logit;
        qacc += logit * logit;

        const float lse = mval + __logf(s);
        zacc += lse * lse;

        if (lr == 0) {                        // one writer per token
            wsProb[token] = 1.0f / s;         // top-1 prob = exp(0)/s
            wsExp[token]  = me;
        }
        out[LOGIT_OFF + (size_t)token * NEXP + lr] = logit;
    }

    // global partial accumulators
    atomicAdd(&ws[WS_PSUM + lr], pacc);       // all 32 lanes, expert lr
    if (lr == 0) atomicAdd(&ws[WS_Z], zacc);  // lanes 0 and 16 (one per token set)
#pragma unroll
    for (int m = 1; m <= 16; m <<= 1) {
        sacc += __shfl_xor(sacc, m, 32);
        qacc += __shfl_xor(qacc, m, 32);
    }
    if (lane == 0) {
        atomicAdd(&ws[WS_LSUM], sacc);
        atomicAdd(&ws[WS_LSQ],  qacc);
    }
}

// ---------------------------------------------------------------------------
// 3) capacity rank via LDS-resident O(N^2) compare, then scatter one-hots.
//    rank(i) = #{j : e_j==e_i && (p_j>p_i || (p_j==p_i && j<i))}
//    survivor iff rank < Ccap; combine weight == 1.0 for top-1 survivors.
// ---------------------------------------------------------------------------
__global__ void rank_scatter(const float* __restrict__ ws_ro,
                             float* __restrict__ ws,
                             float* __restrict__ out) {
    __shared__ float         sp[NTOK];   // 32 KB
    __shared__ unsigned char se[NTOK];   //  8 KB  (fits easily in 320KB WGP LDS)

    const int tid = threadIdx.x;
    const float* prob = ws_ro + WS_PROB;
    const int*   expi = (const int*)ws_ro + WS_EXP;

    for (int i = tid; i < NTOK; i += 512) {
        sp[i] = prob[i];
        se[i] = (unsigned char)expi[i];
    }
    __syncthreads();

    const int i  = blockIdx.x * 512 + tid;
    const float pi = sp[i];
    const int   ei = se[i];

    int rank = 0;
#pragma unroll 8
    for (int j = 0; j < NTOK; ++j) {     // uniform j -> LDS broadcast reads
        const float pj = sp[j];
        const int   ej = se[j];
        if (ej == ei && (pj > pi || (pj == pi && j < i))) ++rank;
    }

    if (rank < CCAP) {
        const size_t off = ((size_t)i * NEXP + ei) * CCAP + rank;
        out[off]            = 1.0f;      // dispatch one-hot
        out[COMB_OFF + off] = 1.0f;      // combine weight (top-1 survivor w==1)
        atomicAdd(&ws[WS_MCNT + ei], 1.0f);
    }
}

// ---------------------------------------------------------------------------
// 4) finalize the three scalars
// ---------------------------------------------------------------------------
__global__ void finalize(const float* __restrict__ ws, float* __restrict__ out) {
    const int lane = threadIdx.x;
    float aux = (lane < NEXP) ? ws[WS_PSUM + lane] * ws[WS_MCNT + lane] : 0.f;
#pragma unroll
    for (int m = 1; m <= 8; m <<= 1) aux += __shfl_xor(aux, m, 32);
    if (lane == 0) {
        const float invN  = 1.0f / (float)NTOK;
        const float invNE = 1.0f / (float)(NTOK * NEXP);
        out[ZLOSS_OFF]     = ws[WS_Z] * invN;                    // z_loss
        out[ZLOSS_OFF + 1] = aux * invN * invN * (float)NEXP;    // aux_loss
        const float mean = ws[WS_LSUM] * invNE;
        const float var  = fmaxf(ws[WS_LSQ] * invNE - mean * mean, 0.0f);
        out[ZLOSS_OFF + 2] = sqrtf(var);                         // logits_std
    }
}

// ---------------------------------------------------------------------------
extern "C" void kernel_launch(void* const* d_in, const int* in_sizes, int n_in,
                              void* d_out, int out_size, void* d_ws, size_t ws_size,
                              hipStream_t stream) {
    const float* X  = (const float*)d_in[0];   // [8192,64,8,8]
    const float* Wg = (const float*)d_in[1];   // [64,16]
    const float* bg = (const float*)d_in[2];   // [16]
    float* out = (float*)d_out;
    float* ws  = (float*)d_ws;

    // 0) zero accumulators
    init_ws<<<1, 64, 0, stream>>>(ws);

    // 1) zero dispatch+combine (671 MB, the bandwidth roofline)
    const long n4 = (long)(2 * COMB_OFF) / 4;   // float4 count
    zero_big<<<2048, 256, 0, stream>>>((v4f*)out, n4);

    // 2) pool + WMMA gate + softmax/argmax/lse (+logits output)
    pool_gate<<<NTOK / 64, 128, 0, stream>>>(X, Wg, bg, out, ws);

    // 3) capacity rank + one-hot scatter
    rank_scatter<<<NTOK / 512, 512, 0, stream>>>(ws, ws, out);

    // 4) scalars
    finalize<<<1, 32, 0, stream>>>(ws, out);
}